// easyLoss_57604101374647
// MI455X (gfx1250) — compile-verified
//
#include <hip/hip_runtime.h>
#include <hip/hip_bf16.h>

// ---------------------------------------------------------------------------
// easyLoss on MI455X (gfx1250, wave32)
// loss_c: streaming BCE over B*A logits (memory-bound: ~17.9MB @ 23.3TB/s)
// loss_e: 1024 gathered EIoU -> per-batch min -> sum  (noise)
// Deterministic: per-block partials in d_ws + single-wave ordered finish.
// ---------------------------------------------------------------------------

#define NBLK      2048            // BCE partial-sum blocks
#define POS_SLOT  NBLK            // ws[NBLK]     : positive-correction partial
#define MIN_BASE  (NBLK + 1)      // ws[NBLK+1..] : per-batch EIoU mins (uint bits)
#define NMIN      256             // >= number of batches (64)
#define WS_WORDS  (NBLK + 1 + NMIN)
#define EPS_F     1e-7f
#define INF_BITS  0x7F800000u

typedef float v2f __attribute__((ext_vector_type(2)));
typedef float v8f __attribute__((ext_vector_type(8)));

// log(1+e^-|x|) via fast hw transcendentals; softplus(x) = max(x,0) + that
__device__ __forceinline__ float softplus_fast(float x) {
    return fmaxf(x, 0.0f) + __logf(1.0f + __expf(-fabsf(x)));
}

__device__ __forceinline__ float wave_reduce_add(float v) {
    #pragma unroll
    for (int off = 16; off > 0; off >>= 1)
        v += __shfl_down(v, off, 32);
    return v;
}

// ---------------------------------------------------------------------------
__global__ void init_ws_kernel(float* __restrict__ ws) {
    int i = blockIdx.x * blockDim.x + threadIdx.x;
    if (i < WS_WORDS) {
        if (i >= MIN_BASE) ((unsigned int*)ws)[i] = INF_BITS;   // +inf mins
        else               ws[i] = 0.0f;                        // zero partials
    }
}

// ---------------------------------------------------------------------------
// Pass 1: treat every element as negative:  sum 0.002 * min(softplus(x), 100)
//   (reference: -0.002 * max(log1p(-sigmoid(x)), -100), log1p(-sig(x)) = -sp)
__global__ void bce_neg_kernel(const float4* __restrict__ x4, int n4,
                               const float* __restrict__ x, int n,
                               float* __restrict__ ws) {
    int tid    = blockIdx.x * blockDim.x + threadIdx.x;
    int stride = gridDim.x * blockDim.x;
    float s = 0.0f;
    for (int i = tid; i < n4; i += stride) {
        float4 v = x4[i];                          // global_load_b128
        s += fminf(softplus_fast(v.x), 100.0f);
        s += fminf(softplus_fast(v.y), 100.0f);
        s += fminf(softplus_fast(v.z), 100.0f);
        s += fminf(softplus_fast(v.w), 100.0f);
    }
    int rem_base = n4 * 4;                          // scalar tail (none here)
    if (tid < n - rem_base) s += fminf(softplus_fast(x[rem_base + tid]), 100.0f);
    s *= 0.002f;

    __shared__ float sm[8];                         // 256 thr = 8 waves
    float w = wave_reduce_add(s);
    int lane = threadIdx.x & 31, wid = threadIdx.x >> 5;
    if (lane == 0) sm[wid] = w;
    __syncthreads();
    if (wid == 0) {
        float t = (lane < (int)(blockDim.x >> 5)) ? sm[lane] : 0.0f;
        t = wave_reduce_add(t);
        if (lane == 0) ws[blockIdx.x] = t;          // fixed slot => deterministic
    }
}

// ---------------------------------------------------------------------------
// Pass 2 (one block, M<=1024 threads): positive-class BCE correction (deduped)
// + decode + EIoU + per-batch atomicMin on uint bits.
__global__ void pos_kernel(const float*  __restrict__ pred_conf,
                           const float4* __restrict__ pred_boxes,
                           const float4* __restrict__ anchors,
                           const float4* __restrict__ gt_boxes,
                           const int*    __restrict__ ai_idx,
                           const int*    __restrict__ batch_idx,
                           int M, int A, float* __restrict__ ws) {
    int j = threadIdx.x;
    float corr = 0.0f;
    if (j < M) {
        int idx  = ai_idx[j];
        bool dup = (j > 0) && (ai_idx[j - 1] == idx);   // .at[].set idempotent
        if (!dup) {
            float xv   = pred_conf[idx];
            float sp   = fmaxf(xv, 0.0f) + log1pf(expf(-fabsf(xv)));
            float lp   = fmaxf(xv - sp, -100.0f);       // log(sigmoid)
            float l1mp = fmaxf(-sp,     -100.0f);       // log(1-sigmoid)
            corr = -lp + 0.002f * l1mp;  // swap neg-weighted term for pos term
        }
        // ---- decode (variances 0.1 / 0.2) ----
        float4 loc = pred_boxes[idx];
        float4 pri = anchors[idx % A];
        float cx = pri.x + loc.x * 0.1f * pri.z;
        float cy = pri.y + loc.y * 0.1f * pri.w;
        float w  = pri.z * expf(loc.z * 0.2f);
        float h  = pri.w * expf(loc.w * 0.2f);
        float b1x1 = cx - 0.5f * w, b1y1 = cy - 0.5f * h;
        float b1x2 = cx + 0.5f * w, b1y2 = cy + 0.5f * h;
        float4 g = gt_boxes[j];
        // ---- EIoU ----
        float ix1 = fmaxf(b1x1, g.x), iy1 = fmaxf(b1y1, g.y);
        float ix2 = fminf(b1x2, g.z), iy2 = fminf(b1y2, g.w);
        float inter = fmaxf(ix2 - ix1, 0.0f) * fmaxf(iy2 - iy1, 0.0f);
        float w1 = b1x2 - b1x1, h1 = b1y2 - b1y1;
        float w2 = g.z - g.x,   h2 = g.w - g.y;
        float uni = w1 * h1 + w2 * h2 - inter + EPS_F;
        float iou = inter / uni;
        float cw = fmaxf(b1x2, g.z) - fminf(b1x1, g.x);
        float ch = fmaxf(b1y2, g.w) - fminf(b1y1, g.y);
        float c2 = cw * cw + ch * ch + EPS_F;
        float dcx = (b1x1 + b1x2) - (g.x + g.z);
        float dcy = (b1y1 + b1y2) - (g.y + g.w);
        float rho2 = 0.25f * (dcx * dcx + dcy * dcy);
        float dw = w1 - w2, dh = h1 - h2;
        float e = 1.0f - iou + rho2 / c2
                + dw * dw / (cw * cw + EPS_F)
                + dh * dh / (ch * ch + EPS_F);
        // e >= 0  =>  uint bit order == float order; min is order-independent
        atomicMin(((unsigned int*)ws) + MIN_BASE + batch_idx[j],
                  __float_as_uint(e));
    }
    __shared__ float sm[32];                        // up to 1024 thr = 32 waves
    float wsum = wave_reduce_add(corr);
    int lane = threadIdx.x & 31, wid = threadIdx.x >> 5;
    if (lane == 0) sm[wid] = wsum;
    __syncthreads();
    if (wid == 0) {
        float t = (lane < (int)(blockDim.x >> 5)) ? sm[lane] : 0.0f;
        t = wave_reduce_add(t);
        if (lane == 0) ws[POS_SLOT] = t;
    }
}

// ---------------------------------------------------------------------------
// Finish: one wave, fixed-order reduction. Cross-lane fold of the 32 per-lane
// partials via V_WMMA_F32_16X16X4_F32: A = ones -> every D element is a column
// sum of B, so sum_lanes(D[0]) = 2 * total (layout-independent, exact f32 FMA).
__global__ void finish_kernel(const float* __restrict__ ws,
                              float* __restrict__ out) {
    int lane = threadIdx.x;                         // 32 threads, EXEC all ones
    float s = 0.0f;
    for (int i = lane; i <= NBLK; i += 32) s += ws[i];   // bce partials + pos

    float tot_c;
#if __has_builtin(__builtin_amdgcn_wmma_f32_16x16x4_f32)
    v2f a; a.x = 1.0f; a.y = 1.0f;                  // A = ones (16x4)
    v2f b; b.x = s;    b.y = 0.0f;                  // B holds the 32 partials
    v8f c = {};
    v8f d = __builtin_amdgcn_wmma_f32_16x16x4_f32(
        false, a, false, b, (short)0, c, false, false);
    tot_c = wave_reduce_add(d[0]) * 0.5f;           // rows 0 & 8 both = colsums
#else
    tot_c = wave_reduce_add(s);
#endif

    float m = 0.0f;
    const unsigned int* mins = ((const unsigned int*)ws) + MIN_BASE;
    for (int i = lane; i < NMIN; i += 32) {
        unsigned int ub = mins[i];
        m += (ub == INF_BITS) ? 0.0f : __uint_as_float(ub);  // isfinite -> 0
    }
    float tot_e = wave_reduce_add(m);

    if (lane == 0) { out[0] = tot_c; out[1] = tot_e; }
}

// ---------------------------------------------------------------------------
extern "C" void kernel_launch(void* const* d_in, const int* in_sizes, int n_in,
                              void* d_out, int out_size, void* d_ws, size_t ws_size,
                              hipStream_t stream) {
    const float*  pred_conf  = (const float*) d_in[0];   // [B*A]
    const float4* pred_boxes = (const float4*)d_in[1];   // [B*A,4]
    const float4* anchors    = (const float4*)d_in[2];   // [A,4]
    const float4* gt_boxes   = (const float4*)d_in[3];   // [M,4]
    const int*    ai_idx     = (const int*)   d_in[4];   // [M]
    const int*    batch_idx  = (const int*)   d_in[5];   // [M]
    float*        out        = (float*)d_out;            // {loss_c, loss_e}
    float*        ws         = (float*)d_ws;

    const int N  = in_sizes[0];          // B*A = 4,480,000
    const int A  = in_sizes[2] / 4;      // 70,000
    const int M  = in_sizes[4];          // 1,024
    const int n4 = N / 4;

    init_ws_kernel<<<(WS_WORDS + 255) / 256, 256, 0, stream>>>(ws);
    bce_neg_kernel<<<NBLK, 256, 0, stream>>>((const float4*)pred_conf, n4,
                                             pred_conf, N, ws);
    pos_kernel<<<1, 1024, 0, stream>>>(pred_conf, pred_boxes, anchors, gt_boxes,
                                       ai_idx, batch_idx, M, A, ws);
    finish_kernel<<<1, 32, 0, stream>>>(ws, out);
}